// ModelNew_84172769067597
// MI455X (gfx1250) — compile-verified
//
#include <hip/hip_runtime.h>
#include <hip/hip_bf16.h>

// C[i, j] = A[i] * B[i, j]   (A: N fp32, B: N x M fp32, row-major)
// Pure streaming kernel: HBM-bound (~512 MB @ 23.3 TB/s => ~23 us floor).
// Strategy: 128-bit vector memory ops, non-temporal hints (no reuse, stream
// exceeds 192 MB L2), one block per row so A[row] is a scalar broadcast.

typedef float v4f __attribute__((ext_vector_type(4)));

// Specialized: M == 4096 (1024 float4 per row), one block of 256 threads per
// row, each thread handles 4 float4s. Loads issued back-to-back first for
// memory-level parallelism, then multiply+store.
__global__ __launch_bounds__(256) void diag_scale_row4096(
    const float* __restrict__ A, const v4f* __restrict__ B4,
    v4f* __restrict__ C4) {
  const int row = blockIdx.x;               // uniform -> scalar path
  const float a = A[row];                   // s_load broadcast
  const size_t base = (size_t)row * 1024 + threadIdx.x;
  const v4f* bp = B4 + base;
  v4f* cp = C4 + base;

  // 4 outstanding global_load_b128 (NT: don't pollute L2, stream-once data)
  v4f b0 = __builtin_nontemporal_load(bp + 0 * 256);
  v4f b1 = __builtin_nontemporal_load(bp + 1 * 256);
  v4f b2 = __builtin_nontemporal_load(bp + 2 * 256);
  v4f b3 = __builtin_nontemporal_load(bp + 3 * 256);

  __builtin_nontemporal_store(b0 * a, cp + 0 * 256);
  __builtin_nontemporal_store(b1 * a, cp + 1 * 256);
  __builtin_nontemporal_store(b2 * a, cp + 2 * 256);
  __builtin_nontemporal_store(b3 * a, cp + 3 * 256);
}

// Generic fallback: any M divisible by 4. One block per row, grid-stride over
// the row's float4s.
__global__ __launch_bounds__(256) void diag_scale_generic(
    const float* __restrict__ A, const v4f* __restrict__ B4,
    v4f* __restrict__ C4, int M4) {
  const int row = blockIdx.x;
  const float a = A[row];
  const size_t base = (size_t)row * (size_t)M4;
  for (int c = threadIdx.x; c < M4; c += blockDim.x) {
    v4f b = __builtin_nontemporal_load(B4 + base + c);
    __builtin_nontemporal_store(b * a, C4 + base + c);
  }
}

// Scalar fallback for odd M (not expected for this problem).
__global__ __launch_bounds__(256) void diag_scale_scalar(
    const float* __restrict__ A, const float* __restrict__ B,
    float* __restrict__ C, int M) {
  const int row = blockIdx.x;
  const float a = A[row];
  const size_t base = (size_t)row * (size_t)M;
  for (int c = threadIdx.x; c < M; c += blockDim.x) {
    C[base + c] = a * B[base + c];
  }
}

extern "C" void kernel_launch(void* const* d_in, const int* in_sizes, int n_in,
                              void* d_out, int out_size, void* d_ws,
                              size_t ws_size, hipStream_t stream) {
  (void)n_in; (void)out_size; (void)d_ws; (void)ws_size;
  const float* A = (const float*)d_in[0];
  const float* B = (const float*)d_in[1];
  float* C = (float*)d_out;

  const int N = in_sizes[0];
  const int M = (N > 0) ? (in_sizes[1] / N) : 0;

  if (M == 4096) {
    diag_scale_row4096<<<N, 256, 0, stream>>>(A, (const v4f*)B, (v4f*)C);
  } else if ((M & 3) == 0) {
    diag_scale_generic<<<N, 256, 0, stream>>>(A, (const v4f*)B, (v4f*)C,
                                              M >> 2);
  } else {
    diag_scale_scalar<<<N, 256, 0, stream>>>(A, B, C, M);
  }
}